// Amend_raf_23519240913227
// MI455X (gfx1250) — compile-verified
//
#include <hip/hip_runtime.h>
#include <hip/hip_bf16.h>

// MI455X / gfx1250 implementation.
//
// Stage 1 (WMMA conv): phase-decomposed depthwise conv as dense f32 WMMA.
//   Taps split ky=8a+r, kx=8b+s.  Per phase r:  M=(a,b) 16 weight taps,
//   K=s (8, two k=4 steps), N=16 image columns (y', j).  Every D entry is a
//   useful partial sum of out[y'-a, j-b]; D accumulates across all 8 phases
//   in registers (16 WMMAs per 16x16 tile, 100% MAC efficiency).  x is read
//   exactly once via non-temporal 8B gathers fully tiling each 512B segment.
//   Scatter goes through a per-workgroup LDS accumulator (ds_add_f32), then
//   one flush of ~1.4K cells/block via global f32 atomics (halo rows overlap
//   between neighboring y-blocks).
// Stage 2: BN folded to affine  xn = s_c*y + t_c:
//   out = 0.5*(s0*y0 + s1*y1) + T + alpha*(G + T),
//   G = (1/BC) sum s_c*y,  T = (1/C) sum t_c.   y (8MB) stays in L2.

typedef __attribute__((ext_vector_type(2))) float v2f;
typedef __attribute__((ext_vector_type(8))) float v8f;

#define HP    125
#define NPIX  (HP * HP)        // 15625
#define NIMG  128              // B*C = 64*2
#define NB    64
#define TOTY  (NIMG * NPIX)    // 2,000,000
#define NOUT  (NB * NPIX)      // 1,000,000

#define LROWS 11               // output rows touched: [Y0-3, Y0+7]
#define LSTR  132              // cols -3..127 (131) padded to 132
#define LSZ   (LROWS * LSTR)   // 1452 floats

__device__ __forceinline__ v8f wmma_f32_k4(v2f a, v2f b, v8f c) {
#if __has_builtin(__builtin_amdgcn_wmma_f32_16x16x4_f32)
  return __builtin_amdgcn_wmma_f32_16x16x4_f32(false, a, false, b,
                                               (short)0, c, false, false);
#else
  asm volatile("v_wmma_f32_16x16x4_f32 %0, %1, %2, %0"
               : "+v"(c) : "v"(a), "v"(b));
  return c;
#endif
}

__device__ __forceinline__ v2f ld_nt_v2f(const float* p) {
  return __builtin_nontemporal_load((const v2f*)p);
}

// ---------------------------------------------------------------- init ----
__global__ __launch_bounds__(256) void k_init(float* __restrict__ ws) {
  int i = blockIdx.x * 256 + threadIdx.x;
  if (i < TOTY + 8) ws[i] = 0.0f;   // y accumulator + stats[4]
}

// ---------------------------------------------------------------- conv ----
// grid = 128 images * 16 y-blocks; block = 256 (8 waves). Wave w handles
// y' = 8*yb + w; loops over 8 j-tiles of 16 columns each.
__global__ __launch_bounds__(256) void k_conv_wmma(
    const float* __restrict__ x, const float* __restrict__ w,
    float* __restrict__ y) {
  const int img  = blockIdx.x >> 4;        // 0..127
  const int yb   = blockIdx.x & 15;        // 0..15
  const int wave = threadIdx.x >> 5;
  const int lane = threadIdx.x & 31;
  const int yp   = yb * 8 + wave;          // y' in 0..127
  const int n    = lane & 15;              // N column slot
  const int kh   = lane >> 4;              // lane-half: K slots {2kh, 2kh+1}
  const int aa   = (lane & 15) >> 2;       // A row slot m=(a,b)
  const int bb   = lane & 3;

  __shared__ float lacc[LSZ];
  for (int i = threadIdx.x; i < LSZ; i += 256) lacc[i] = 0.0f;
  __syncthreads();

  // Preload all A operands: A[r][step], tap s = 4*step + 2*kh + {0,1}
  v2f A[8][2];
#pragma unroll
  for (int r = 0; r < 8; ++r)
#pragma unroll
    for (int st = 0; st < 2; ++st) {
      int s0 = 4 * st + 2 * kh;
      A[r][st] = *(const v2f*)(w + (8 * aa + r) * 32 + 8 * bb + s0);
    }

  const float* ximg = x + (size_t)img * (1024u * 1024u);

  for (int j0 = 0; j0 < 128; j0 += 16) {
    // prefetch next j-tile's 8 row segments (512B each, spread over lanes)
    if (j0 < 112) {
#pragma unroll
      for (int r = 0; r < 8; ++r)
        __builtin_prefetch(ximg + (size_t)(8 * yp + r) * 1024 +
                               8 * (j0 + 16) + lane * 4, 0, 1);
    }
    v8f acc = {};
#pragma unroll
    for (int r = 0; r < 8; ++r) {
      const float* xrow = ximg + (size_t)(8 * yp + r) * 1024;
#pragma unroll
      for (int st = 0; st < 2; ++st) {
        int s0 = 4 * st + 2 * kh;
        v2f B = ld_nt_v2f(xrow + 8 * (j0 + n) + s0);  // 8B NT gather
        acc = wmma_f32_k4(A[r][st], B, acc);
      }
    }
    // Scatter D into LDS: VGPR v holds row m' = v + 8*kh -> (a,b);
    // col n -> (y', j0+n).  li/ci are always in-bounds (no predicates).
#pragma unroll
    for (int v = 0; v < 8; ++v) {
      int mm = v + 8 * kh;
      int av = mm >> 2, bv = mm & 3;
      int li = wave + 3 - av;          // 0..10
      int ci = j0 + n - bv + 3;        // 0..130
      atomicAdd(&lacc[li * LSTR + ci], acc[v]);
    }
  }
  __syncthreads();

  // Flush LDS accumulator to global (halo rows overlap across y-blocks).
  float* yimg = y + (size_t)img * NPIX;
  const int Y0 = yb * 8;
  for (int k = threadIdx.x; k < LSZ; k += 256) {
    int li = k / LSTR;
    int ci = k - li * LSTR;
    int oy = Y0 - 3 + li;
    int ox = ci - 3;
    if (oy >= 0 && oy < HP && ox >= 0 && ox < HP)
      atomicAdd(yimg + oy * HP + ox, lacc[k]);
  }
}

// --------------------------------------------------------------- stats ----
// One block per image; y is L2-resident.
__global__ __launch_bounds__(256) void k_stats(const float* __restrict__ y,
                                               float* __restrict__ stats) {
  const int img = blockIdx.x;              // 0..127, c = img & 1
  const float* yi = y + (size_t)img * NPIX;
  float s = 0.0f, q = 0.0f;
  for (int p = threadIdx.x; p < NPIX; p += 256) {
    float v = yi[p];
    s += v; q += v * v;
  }
  __shared__ float sh[512];
  int t = threadIdx.x;
  sh[t] = s; sh[256 + t] = q;
  __syncthreads();
  for (int off = 128; off > 0; off >>= 1) {
    if (t < off) { sh[t] += sh[t + off]; sh[256 + t] += sh[256 + t + off]; }
    __syncthreads();
  }
  if (t == 0) {
    int c = img & 1;
    atomicAdd(&stats[2 * c + 0], sh[0]);
    atomicAdd(&stats[2 * c + 1], sh[256]);
  }
}

// ------------------------------------------------------------ finalize ----
__global__ void k_finalize(const float* __restrict__ stats,
                           const float* __restrict__ gamma,
                           const float* __restrict__ beta,
                           float* __restrict__ coef) {
  const float Ninv = 1.0f / (float)(NB * NPIX);
  float t[2];
  for (int c = 0; c < 2; ++c) {
    float mean = stats[2 * c + 0] * Ninv;
    float var  = stats[2 * c + 1] * Ninv - mean * mean;
    float s    = gamma[c] * rsqrtf(var + 1e-5f);
    t[c]       = beta[c] - mean * s;
    coef[c]    = s;
  }
  coef[2] = 0.5f * (t[0] + t[1]);  // T
}

// ------------------------------------------------------------------- G ----
__global__ __launch_bounds__(256) void k_gmean(const float* __restrict__ y,
                                               const float* __restrict__ coef,
                                               float* __restrict__ G) {
  int p = blockIdx.x * 256 + threadIdx.x;
  if (p >= NPIX) return;
  float s0 = coef[0], s1 = coef[1];
  float acc = 0.0f;
  for (int img = 0; img < NIMG; img += 2) {
    acc += s0 * y[(size_t)img * NPIX + p];
    acc += s1 * y[(size_t)(img + 1) * NPIX + p];
  }
  G[p] = acc * (1.0f / (float)NIMG);
}

// ----------------------------------------------------------------- out ----
__global__ __launch_bounds__(256) void k_out(const float* __restrict__ y,
                                             const float* __restrict__ coef,
                                             const float* __restrict__ G,
                                             const float* __restrict__ alpha,
                                             float* __restrict__ out) {
  int i = blockIdx.x * 256 + threadIdx.x;
  if (i >= NOUT) return;
  int b = i / NPIX;
  int p = i - b * NPIX;
  float s0 = coef[0], s1 = coef[1], T = coef[2];
  float al = alpha[0];
  float xm = 0.5f * (s0 * y[(size_t)(2 * b) * NPIX + p] +
                     s1 * y[(size_t)(2 * b + 1) * NPIX + p]) + T;
  out[i] = xm + al * (G[p] + T);
  if (i == 0) out[NOUT] = al;  // second tuple element (alpha)
}

// -------------------------------------------------------------- launch ----
extern "C" void kernel_launch(void* const* d_in, const int* in_sizes, int n_in,
                              void* d_out, int out_size, void* d_ws,
                              size_t ws_size, hipStream_t stream) {
  const float* x     = (const float*)d_in[0];
  const float* w     = (const float*)d_in[1];
  const float* gamma = (const float*)d_in[2];
  const float* beta  = (const float*)d_in[3];
  const float* alpha = (const float*)d_in[4];
  float* out = (float*)d_out;

  float* wsf   = (float*)d_ws;
  float* y     = wsf;                  // 2,000,000
  float* stats = wsf + TOTY;           // 4
  float* coef  = wsf + TOTY + 8;       // 3
  float* G     = wsf + TOTY + 16;      // 15625

  k_init<<<(TOTY + 8 + 255) / 256, 256, 0, stream>>>(wsf);
  k_conv_wmma<<<128 * 16, 256, 0, stream>>>(x, w, y);
  k_stats<<<NIMG, 256, 0, stream>>>(y, stats);
  k_finalize<<<1, 1, 0, stream>>>(stats, gamma, beta, coef);
  k_gmean<<<(NPIX + 255) / 256, 256, 0, stream>>>(y, coef, G);
  k_out<<<(NOUT + 255) / 256, 256, 0, stream>>>(y, coef, G, alpha, out);
}